// ConditionalRandomField_35562329211832
// MI455X (gfx1250) — compile-verified
//
#include <hip/hip_runtime.h>
#include <hip/hip_bf16.h>
#include <stdint.h>

#define CRF_B 512
#define CRF_S 512
#define CRF_T 128
#define CRF_N (CRF_B * CRF_S)

typedef __attribute__((ext_vector_type(2))) float    v2f;
typedef __attribute__((ext_vector_type(8))) float    v8f;
typedef __attribute__((ext_vector_type(4))) uint32_t u32x4;
typedef __attribute__((ext_vector_type(4))) int32_t  i32x4;
typedef __attribute__((ext_vector_type(8))) int32_t  i32x8;

// Address-space-qualified vector pointer types for the async-LDS builtins
// (probe-confirmed prototype: (addrspace(1) v4i*, addrspace(3) v4i*, Ii, Ii)).
typedef __attribute__((address_space(1))) i32x4 as1_i32x4;
typedef __attribute__((address_space(3))) i32x4 as3_i32x4;

// ---------------------------------------------------------------------------
// Wave32 sum reduction via V_WMMA_F32_16X16X4_F32 with A = ones:
//   D[m][n] = sum_k A[m][k]*B[k][n] + 0 = sum_k B[k][n]   (column sums)
// Each lane's partial goes into B (vgpr0), 0 into B (vgpr1); then
// sum_n colsum[n] == sum of all 32 partials, independent of the exact
// K/N lane striping. D vgpr0 holds colsum[lane%16] in every lane, so a
// 4-step butterfly over 16 lanes yields the wave total (exact in f32,
// since A elements are exactly 1.0f).
// ---------------------------------------------------------------------------
__device__ __forceinline__ float wave_sum32(float x) {
#if __has_builtin(__builtin_amdgcn_wmma_f32_16x16x4_f32)
  v2f a; a.x = 1.0f; a.y = 1.0f;   // A = ones(16x4)
  v2f b; b.x = x;    b.y = 0.0f;   // B carries the 32 partials
  v8f c = {};
  v8f d = __builtin_amdgcn_wmma_f32_16x16x4_f32(
      /*neg_a=*/false, a, /*neg_b=*/false, b,
      /*c_mod=*/(short)0, c, /*reuse_a=*/false, /*reuse_b=*/false);
  float s = d[0];
  s += __shfl_xor(s, 1, 32);
  s += __shfl_xor(s, 2, 32);
  s += __shfl_xor(s, 4, 32);
  s += __shfl_xor(s, 8, 32);
  return s;
#else
  for (int off = 16; off; off >>= 1) x += __shfl_xor(x, off, 32);
  return x;
#endif
}

// ---------------------------------------------------------------------------
// Stage the 128x128 f32 transitions table (64 KB) into LDS.
// Tier 1: Tensor Data Mover descriptor load (6-arg or 5-arg builtin form)
// Tier 2: async global->LDS loads (ASYNCcnt)
// Tier 3: plain vector loads + LDS stores
// ---------------------------------------------------------------------------
__device__ __forceinline__ void stage_transitions(const float* __restrict__ g,
                                                  float* s_trans) {
#if __has_builtin(__builtin_amdgcn_tensor_load_to_lds)
  if (threadIdx.x < 32) {           // one wave issues the DMA (all 32 lanes live)
    uint64_t ga  = (uint64_t)(uintptr_t)g;
    uint32_t lds = (uint32_t)(uintptr_t)s_trans;  // low 32 bits = LDS byte offset

    // D# group 0: count=1 | lds_addr | global_addr[56:0] | type=2
    u32x4 g0;
    g0.x = 0x1u;
    g0.y = lds;
    g0.z = (uint32_t)(ga & 0xFFFFFFFFu);
    g0.w = ((uint32_t)(ga >> 32) & 0x01FFFFFFu) | (2u << 30);

    // D# group 1: data_size=4B, tensor 128x128 stride 128, tile 128x128
    i32x8 g1;
    g1[0] = (int32_t)0x00020000u;       // workgroup_mask=0, data_size=2 (4B)
    g1[1] = (int32_t)(128u << 16);      // tensor_dim0[15:0]=128 (bits 79:48)
    g1[2] = (int32_t)(128u << 16);      // tensor_dim0 hi=0 | tensor_dim1[15:0]=128
    g1[3] = (int32_t)(128u << 16);      // tensor_dim1 hi=0 | tile_dim0=128
    g1[4] = (int32_t)128;               // tile_dim1=128, tile_dim2=0
    g1[5] = (int32_t)128;               // tensor_dim0_stride[31:0]=128
    g1[6] = 0;                          // stride hi=0, tensor_dim1_stride=0 (2D)
    g1[7] = 0;

    i32x4 z4 = {};
#if __has_include(<hip/amd_detail/amd_gfx1250_TDM.h>)
    // amdgpu-toolchain (clang-23 / therock headers): 6-arg builtin form
    i32x8 z8 = {};
    __builtin_amdgcn_tensor_load_to_lds(g0, g1, z4, z4, z8, 0);
#else
    // ROCm 7.2 (clang-22): 5-arg builtin form
    __builtin_amdgcn_tensor_load_to_lds(g0, g1, z4, z4, 0);
#endif
  }
  __builtin_amdgcn_s_wait_tensorcnt(0);
  __syncthreads();
#elif __has_builtin(__builtin_amdgcn_global_load_async_to_lds_b128)
  // ---- async global->LDS path (ASYNCcnt) ----
  for (int i = threadIdx.x * 4; i < CRF_T * CRF_T; i += (int)blockDim.x * 4) {
    __builtin_amdgcn_global_load_async_to_lds_b128(
        (as1_i32x4*)(g + i),        // global src (addrspace(1) v4i*)
        (as3_i32x4*)(s_trans + i),  // LDS dst    (addrspace(3) v4i*)
        /*offset=*/0, /*cpol=*/0);
  }
#if __has_builtin(__builtin_amdgcn_s_wait_asynccnt)
  __builtin_amdgcn_s_wait_asynccnt(0);
#else
  asm volatile("s_wait_asynccnt 0x0" ::: "memory");
#endif
  __syncthreads();
#else
  // ---- portable fallback ----
  for (int i = threadIdx.x * 4; i < CRF_T * CRF_T; i += (int)blockDim.x * 4) {
    float4 v = *(const float4*)(g + i);
    *(float4*)(s_trans + i) = v;
  }
  __syncthreads();
#endif
}

// ---------------------------------------------------------------------------
// Kernels
// ---------------------------------------------------------------------------
__global__ void crf_zero_kernel(float* acc) {
  acc[0] = 0.0f;   // score accumulator
  acc[1] = 0.0f;   // mask count accumulator
}

__global__ __launch_bounds__(256)
void crf_score_kernel(const float* __restrict__ emissions,
                      const int* __restrict__ tags,
                      const unsigned char* __restrict__ mask,
                      const float* __restrict__ transitions,
                      float* __restrict__ acc) {
  __shared__ float s_trans[CRF_T * CRF_T];   // 64 KB
  stage_transitions(transitions, s_trans);

  float score = 0.0f;
  float cnt   = 0.0f;

  // 512 blocks * 256 threads = 131072 threads; N = 262144 -> exactly 2
  // iterations per thread, zero divergence (EXEC all-1s at the WMMA below).
  for (int idx = (int)blockIdx.x * (int)blockDim.x + (int)threadIdx.x;
       idx < CRF_N;
       idx += (int)gridDim.x * (int)blockDim.x) {
    const int s  = idx & (CRF_S - 1);
    const int t0 = tags[idx];
    const float m0 = mask[idx] ? 1.0f : 0.0f;

    // emission gather: one 4B element out of a 512B row
    const float e = emissions[(size_t)idx * CRF_T + t0];
    score += e * m0;
    cnt   += m0;

    if (s < CRF_S - 1) {
      const int   t1 = tags[idx + 1];
      const float m1 = mask[idx + 1] ? 1.0f : 0.0f;
      score += s_trans[t0 * CRF_T + t1] * (m0 * m1);
    }
  }

  const float wscore = wave_sum32(score);
  const float wcnt   = wave_sum32(cnt);
  if ((threadIdx.x & 31u) == 0u) {
    atomicAdd(&acc[0], wscore);
    atomicAdd(&acc[1], wcnt);
  }
}

__global__ void crf_final_kernel(const float* __restrict__ acc,
                                 float* __restrict__ out) {
  out[0] = acc[0] / acc[1];
}

// ---------------------------------------------------------------------------
// Launch
// ---------------------------------------------------------------------------
extern "C" void kernel_launch(void* const* d_in, const int* in_sizes, int n_in,
                              void* d_out, int out_size, void* d_ws, size_t ws_size,
                              hipStream_t stream) {
  const float*         emissions   = (const float*)d_in[0];
  const int*           tags        = (const int*)d_in[1];           // int64 -> int32 canonical
  const unsigned char* mask        = (const unsigned char*)d_in[2]; // jnp.bool_ = 1 byte
  const float*         transitions = (const float*)d_in[3];
  float* out = (float*)d_out;
  float* acc = (float*)d_ws;   // [0]=score, [1]=count

  crf_zero_kernel<<<1, 1, 0, stream>>>(acc);
  crf_score_kernel<<<512, 256, 0, stream>>>(emissions, tags, mask, transitions, acc);
  crf_final_kernel<<<1, 1, 0, stream>>>(acc, out);
  (void)in_sizes; (void)n_in; (void)out_size; (void)ws_size;
}